// SampleConcrete_20486994002397
// MI455X (gfx1250) — compile-verified
//
#include <hip/hip_runtime.h>
#include <stdint.h>

// Problem geometry (from the reference): B x SLEN, top-k threshold with K=256,
// TAU = 0.5 (so 1/TAU = 2).
#define SLEN      8192
#define K_SEL     256
#define NTHREADS  256                  // 8 wave32 waves per workgroup
#define EPT       (SLEN / NTHREADS)    // 32 elements per thread
#define V4PT      (EPT / 4)            // 8 float4 chunks per thread

// Native clang vector (not HIP_vector_type) so nontemporal builtins accept it.
typedef float v4f __attribute__((ext_vector_type(4)));

// Monotonic float -> u32 key: larger float <=> larger key (no NaNs in input).
__device__ __forceinline__ uint32_t monokey(float f) {
  uint32_t u = __float_as_uint(f);
  return (u & 0x80000000u) ? ~u : (u | 0x80000000u);
}

// Byte offset of an LDS object within the workgroup's LDS allocation.
// AS(3) pointers are 32-bit offsets on amdgcn; this is what the async-to-LDS
// instruction's VDST operand wants.
__device__ __forceinline__ uint32_t lds_off_u32(const void* p) {
  return (uint32_t)(uintptr_t)(__attribute__((address_space(3))) const void*)p;
}

__device__ __forceinline__ void concrete1(float lg, float ua, float ub,
                                          uint32_t thrKey, float& s, float& d) {
  // g = -log(-log(u));  samples = n/(n+m) with
  //   n = exp(2*(g_a + lg)), m = exp(2*(g_b + 1 - lg))
  // computed stably as 1/(1 + exp(arg_m - arg_n)): one exp, one rcp.
  const float ga = -__logf(-__logf(ua));
  const float gb = -__logf(-__logf(ub));
  const float x  = 2.0f * ((gb + 1.0f - lg) - (ga + lg));
  s = 1.0f / (1.0f + __expf(x));
  d = (monokey(lg) >= thrKey) ? 1.0f : 0.0f;
}

__global__ __launch_bounds__(NTHREADS)
void sample_concrete_topk(const float* __restrict__ logits,
                          const float* __restrict__ unifA,
                          const float* __restrict__ unifB,
                          float* __restrict__ outS,
                          float* __restrict__ outD) {
  __shared__ __align__(16) float s_lg[SLEN];   // 32 KB: one logits row
  __shared__ uint32_t s_hist[256];             // radix histogram
  __shared__ uint32_t s_prefix;                // selected high bits so far
  __shared__ uint32_t s_krem;                  // remaining rank within prefix

  const uint32_t row = blockIdx.x;
  const uint32_t tid = threadIdx.x;
  const float* rowL = logits + (size_t)row * SLEN;
  const float* rowA = unifA  + (size_t)row * SLEN;
  const float* rowB = unifB  + (size_t)row * SLEN;

  // ---- Phase A: CDNA5 async DMA of the logits row into LDS (ASYNCcnt path).
  {
    const uint32_t lbase = lds_off_u32(&s_lg[0]);
#pragma unroll
    for (int i = 0; i < V4PT; ++i) {
      const uint32_t c = (uint32_t)i * NTHREADS + tid;            // float4 chunk
      const uint64_t g = (uint64_t)(uintptr_t)(rowL + (size_t)c * 4u);
      const uint32_t l = lbase + c * 16u;
      asm volatile("global_load_async_to_lds_b128 %0, %1, off"
                   :: "v"(l), "v"(g) : "memory");
    }
  }

  // While the DMA is in flight: prefetch this block's uniforms into L2
  // (consumed in phase C, after the LDS-only radix select).
#pragma unroll
  for (int i = 0; i < V4PT; ++i) {
    const uint32_t c = (uint32_t)i * NTHREADS + tid;
    __builtin_prefetch(rowA + (size_t)c * 4u, 0, 1);
    __builtin_prefetch(rowB + (size_t)c * 4u, 0, 1);
  }

  if (tid == 0) { s_prefix = 0u; s_krem = K_SEL; }

  asm volatile("s_wait_asynccnt 0x0" ::: "memory");
  __syncthreads();   // all waves' DMA complete + s_prefix/s_krem visible

  // ---- Phase B: 4-pass MSB-first radix select of the K-th largest key.
#pragma unroll 1
  for (int pass = 0; pass < 4; ++pass) {
    const int shift = 24 - 8 * pass;
    s_hist[tid] = 0u;
    __syncthreads();
    const uint32_t pfx = s_prefix;
#pragma unroll
    for (int i = 0; i < EPT; ++i) {
      const uint32_t key = monokey(s_lg[(uint32_t)i * NTHREADS + tid]);
      // bits above the current digit must match the prefix found so far
      // (u64 shift so pass 0's shift of 32 is well-defined -> 0 == 0).
      const uint32_t hi = (uint32_t)((uint64_t)key >> (uint32_t)(shift + 8));
      if (hi == pfx)
        atomicAdd(&s_hist[(key >> shift) & 0xFFu], 1u);   // ds_add_u32
    }
    __syncthreads();
    if (tid == 0) {
      const uint32_t k = s_krem;
      uint32_t cum = 0u;
      for (int b = 255; b >= 0; --b) {
        const uint32_t c = s_hist[b];
        if (cum + c >= k) {
          s_krem   = k - cum;
          s_prefix = (s_prefix << 8) | (uint32_t)b;
          break;
        }
        cum += c;
      }
    }
    __syncthreads();
  }
  const uint32_t thrKey = s_prefix;   // key of the exact K-th largest value

  // ---- Phase C: elementwise sampling + thresholding, 16B vector in/out.
#pragma unroll 1
  for (int i = 0; i < V4PT; ++i) {
    const uint32_t c   = (uint32_t)i * NTHREADS + tid;
    const uint32_t idx = c * 4u;
    const v4f lg4 = *reinterpret_cast<const v4f*>(&s_lg[idx]);
    const v4f a4  = *reinterpret_cast<const v4f*>(rowA + idx);
    const v4f b4  = *reinterpret_cast<const v4f*>(rowB + idx);
    v4f s4, d4;
#pragma unroll
    for (int k = 0; k < 4; ++k) {
      float sv, dv;
      concrete1(lg4[k], a4[k], b4[k], thrKey, sv, dv);
      s4[k] = sv;
      d4[k] = dv;
    }
    // Outputs are never re-read: non-temporal stores keep L2 for the uniforms.
    __builtin_nontemporal_store(s4, reinterpret_cast<v4f*>(outS + (size_t)row * SLEN + idx));
    __builtin_nontemporal_store(d4, reinterpret_cast<v4f*>(outD + (size_t)row * SLEN + idx));
  }
}

extern "C" void kernel_launch(void* const* d_in, const int* in_sizes, int n_in,
                              void* d_out, int out_size, void* d_ws, size_t ws_size,
                              hipStream_t stream) {
  const float* logits = (const float*)d_in[0];   // (B, SLEN, 1) f32
  const float* ua     = (const float*)d_in[1];   // (B, 1, SLEN) f32
  const float* ub     = (const float*)d_in[2];   // (B, 1, SLEN) f32
  float* out = (float*)d_out;                    // samples | discrete, concat

  const int Brows = in_sizes[0] / SLEN;          // 1024
  float* outS = out;
  float* outD = out + (size_t)Brows * SLEN;

  hipLaunchKernelGGL(sample_concrete_topk, dim3(Brows), dim3(NTHREADS), 0, stream,
                     logits, ua, ub, outS, outD);
}